// AttentionDecoder_87995289960566
// MI455X (gfx1250) — compile-verified
//
#include <hip/hip_runtime.h>
#include <math.h>

typedef __attribute__((ext_vector_type(2))) float v2f;
typedef __attribute__((ext_vector_type(8))) float v8f;

#define DB 64      // batch
#define DS 196     // feature seq len
#define DT 16      // caption len
#define DH 1024    // hidden
#define DK 2048    // keys hidden
#define DE 512     // embed
#define DV 10000   // vocab
#define NSTEP 15   // T-1 decode steps

// Flat shared-pointer low 32 bits == LDS byte offset (ISA 10.2: LDS aperture
// addresses truncate to addr[31:0] for the LDS address).
__device__ __forceinline__ unsigned lds_off(const void* p) {
    return (unsigned)(size_t)p;
}

// ---------------------------------------------------------------------------
// Generic fp32 WMMA GEMM: C[M,N] = A[M,K] * B + bias, B optionally transposed
// (transB: B element (k,n) = Bm[n*ldb + k], i.e. computes A @ W^T for row-major W).
// Requirements: M % 64 == 0 (grid.y = M/64), K % 32 == 0, N % 4 == 0 (ragged in
// units of 4 OK via guards). Block = 256 threads (8 waves), 64x64 C tile.
// Wave w: rows (w&3)*16..+15, cols (w>>2)*32..+31 (two 16x16 WMMA tiles).
// A tile staged via GLOBAL_LOAD_ASYNC_TO_LDS_B128 (ASYNCcnt); B tile staged
// transposed (Bs[n][k]) so WMMA fragments are contiguous float2 ds_load_b64.
// ---------------------------------------------------------------------------
__global__ __launch_bounds__(256)
void gemm_wmma_f32(const float* __restrict__ A, int lda,
                   const float* __restrict__ Bm, int ldb, int transB,
                   const float* __restrict__ bias,
                   float* __restrict__ C, int ldc,
                   int N, int Kdim)
{
    __shared__ __align__(16) float As[64][36];   // [m][k], stride 36 -> conflict-free frags
    __shared__ __align__(16) float Bs[64][36];   // [n][k] (transposed tile)

    const int t    = threadIdx.x;
    const int m0   = blockIdx.y * 64;
    const int n0   = blockIdx.x * 64;
    const int lane = t & 31;
    const int wv   = t >> 5;
    const int l    = lane & 15;
    const int hi   = lane >> 4;             // lane half selects K pair {0,1}/{2,3}
    const int m_off = (wv & 3) * 16;
    const int n_off = (wv >> 2) * 32;

    v8f acc0 = {};
    v8f acc1 = {};

    for (int k0 = 0; k0 < Kdim; k0 += 32) {
        // ---- stage A 64x32: async global->LDS, 16B per lane, no VGPR round-trip ----
        #pragma unroll
        for (int i = 0; i < 2; ++i) {
            int idx4 = t * 2 + i;
            int row  = idx4 >> 3;
            int c4   = (idx4 & 7) * 4;
            unsigned            lp = lds_off(&As[row][c4]);
            unsigned long long  gp = (unsigned long long)(const void*)
                                     (A + (size_t)(m0 + row) * lda + k0 + c4);
            asm volatile("global_load_async_to_lds_b128 %0, %1, off"
                         :: "v"(lp), "v"(gp) : "memory");
        }
        // ---- stage B 32x64 into Bs[n][k] ----
        if (transB) {
            // memory is k-contiguous per n row: float4 along k, direct store
            #pragma unroll
            for (int i = 0; i < 2; ++i) {
                int e  = t + i * 256;       // 0..511 float4s
                int nn = e >> 3;
                int k4 = (e & 7) * 4;
                int gn = n0 + nn;
                float4 vb = make_float4(0.f, 0.f, 0.f, 0.f);
                if (gn < N)
                    vb = *(const float4*)(Bm + (size_t)gn * ldb + k0 + k4);
                *(float4*)&Bs[nn][k4] = vb;
            }
        } else {
            // memory is n-contiguous per k row: float4 along n, transpose on store
            #pragma unroll
            for (int i = 0; i < 2; ++i) {
                int e  = t + i * 256;       // 0..511 float4s
                int kk = e >> 4;
                int n4 = (e & 15) * 4;
                int gn = n0 + n4;
                float4 vb = make_float4(0.f, 0.f, 0.f, 0.f);
                if (gn < N)                 // N % 4 == 0 -> whole float4 in range
                    vb = *(const float4*)(Bm + (size_t)(k0 + kk) * ldb + gn);
                Bs[n4 + 0][kk] = vb.x;
                Bs[n4 + 1][kk] = vb.y;
                Bs[n4 + 2][kk] = vb.z;
                Bs[n4 + 3][kk] = vb.w;
            }
        }
        asm volatile("s_wait_asynccnt 0x0" ::: "memory");
        __syncthreads();

        // ---- 8 x V_WMMA_F32_16X16X4_F32 per accumulator; contiguous b64 frags ----
        #pragma unroll
        for (int kk = 0; kk < 8; ++kk) {
            const int kb = kk * 4 + 2 * hi; // lanes 0-15: K={0,1}, lanes 16-31: K={2,3}
            v2f a  = *(const v2f*)&As[m_off + l][kb];
            v2f b0 = *(const v2f*)&Bs[n_off + l][kb];
            v2f b1 = *(const v2f*)&Bs[n_off + 16 + l][kb];
            acc0 = __builtin_amdgcn_wmma_f32_16x16x4_f32(
                       false, a, false, b0, (short)0, acc0, false, false);
            acc1 = __builtin_amdgcn_wmma_f32_16x16x4_f32(
                       false, a, false, b1, (short)0, acc1, false, false);
        }
        __syncthreads();
    }

    // ---- store (C/D layout: VGPR r -> M=r (lanes 0-15) / M=r+8 (lanes 16-31)) ----
    #pragma unroll
    for (int r = 0; r < 8; ++r) {
        int row = m0 + m_off + r + hi * 8;
        int c0  = n0 + n_off + l;
        int c1  = c0 + 16;
        float v0 = acc0[r];
        float v1 = acc1[r];
        if (c0 < N) {
            if (bias) v0 += bias[c0];
            C[(size_t)row * ldc + c0] = v0;
        }
        if (c1 < N) {
            if (bias) v1 += bias[c1];
            C[(size_t)row * ldc + c1] = v1;
        }
    }
}

// ---------------------------------------------------------------------------
// Gather embedding rows for teacher-forcing tokens: token[b,0]=sos, else captions[b,j]
// ---------------------------------------------------------------------------
__global__ void embed_gather(const float* __restrict__ emb,
                             const int* __restrict__ captions,
                             const int* __restrict__ sos,
                             float* __restrict__ xemb)
{
    int m = blockIdx.x;                 // 0 .. B*NSTEP-1
    int b = m / NSTEP, j = m % NSTEP;
    int tok = (j == 0) ? sos[0] : captions[b * DT + j];
    const float4* src = (const float4*)(emb + (size_t)tok * DE);
    float4* dst = (float4*)(xemb + (size_t)m * DE);
    dst[threadIdx.x] = src[threadIdx.x];
}

// ---------------------------------------------------------------------------
// e[b,s] = v . tanh(q[b] + keys_proj[b,s]) + bv ; w[b,s] = softmax_s(e)
// ---------------------------------------------------------------------------
__global__ __launch_bounds__(256)
void attn_scores(const float* __restrict__ q, const float* __restrict__ kp,
                 const float* __restrict__ v, const float* __restrict__ bv,
                 float* __restrict__ w)
{
    __shared__ float qs[DH];
    __shared__ float vs[DH];
    __shared__ float es[DS];
    __shared__ float red[256];
    const int b = blockIdx.x, t = threadIdx.x;
    for (int i = t; i < DH; i += 256) { qs[i] = q[(size_t)b * DH + i]; vs[i] = v[i]; }
    __syncthreads();

    const int lane = t & 31, wv = t >> 5;
    const float bvv = bv[0];
    for (int s = wv; s < DS; s += 8) {
        const float* kr = kp + ((size_t)b * DS + s) * DH;
        float acc = 0.f;
        for (int i = lane; i < DH; i += 32)
            acc += tanhf(qs[i] + kr[i]) * vs[i];
        for (int off = 16; off > 0; off >>= 1)
            acc += __shfl_down(acc, off, 32);
        if (lane == 0) es[s] = acc + bvv;
    }
    __syncthreads();

    red[t] = (t < DS) ? es[t] : -3.4e38f;
    __syncthreads();
    for (int o = 128; o > 0; o >>= 1) {
        if (t < o) red[t] = fmaxf(red[t], red[t + o]);
        __syncthreads();
    }
    const float mx = red[0];
    __syncthreads();
    const float ex = (t < DS) ? __expf(es[t] - mx) : 0.f;
    red[t] = ex;
    __syncthreads();
    for (int o = 128; o > 0; o >>= 1) {
        if (t < o) red[t] += red[t + o];
        __syncthreads();
    }
    const float inv = 1.f / red[0];
    if (t < DS) w[(size_t)b * DS + t] = ex * inv;
}

// ---------------------------------------------------------------------------
// ctx_raw[b,k] = sum_s w[b,s] * features[b,s,k]
// ---------------------------------------------------------------------------
__global__ __launch_bounds__(256)
void ctx_sum(const float* __restrict__ w, const float* __restrict__ feat,
             float* __restrict__ ctxr)
{
    __shared__ float ws[DS];
    const int b = blockIdx.x, t = threadIdx.x;
    for (int i = t; i < DS; i += 256) ws[i] = w[(size_t)b * DS + i];
    __syncthreads();
    float acc[8] = {0, 0, 0, 0, 0, 0, 0, 0};
    const float* fb = feat + (size_t)b * DS * DK;
    for (int s = 0; s < DS; ++s) {
        const float wb = ws[s];
        const float* fr = fb + (size_t)s * DK;
        #pragma unroll
        for (int j = 0; j < 8; ++j) acc[j] += wb * fr[t + 256 * j];
    }
    #pragma unroll
    for (int j = 0; j < 8; ++j) ctxr[(size_t)b * DK + t + 256 * j] = acc[j];
}

// xc[b, 0:H] = x_seq[b, tstep, :];  xc[b, H:2H] = ctx[b, :]
__global__ void concat_x(const float* __restrict__ xseq, const float* __restrict__ ctx,
                         float* __restrict__ xc, int tstep)
{
    int idx = blockIdx.x * 256 + threadIdx.x;   // < DB*DK
    int b = idx >> 11, c = idx & 2047;
    xc[idx] = (c < DH) ? xseq[((size_t)b * NSTEP + tstep) * DH + c]
                       : ctx[(size_t)b * DH + (c - DH)];
}

// PyTorch GRU cell gate fusion (r,z,n order); updates h in place
__global__ void gru_fuse(const float* __restrict__ gi, const float* __restrict__ gh,
                         float* __restrict__ h)
{
    int idx = blockIdx.x * 256 + threadIdx.x;   // < DB*DH
    int b = idx >> 10, n = idx & 1023;
    size_t base = (size_t)b * 3 * DH;
    float ir = gi[base + n], iz = gi[base + DH + n], inn = gi[base + 2 * DH + n];
    float hr = gh[base + n], hz = gh[base + DH + n], hn  = gh[base + 2 * DH + n];
    float r  = 1.f / (1.f + __expf(-(ir + hr)));
    float z  = 1.f / (1.f + __expf(-(iz + hz)));
    float nn = tanhf(inn + r * hn);
    float ho = h[idx];
    h[idx] = (1.f - z) * nn + z * ho;
}

// ---------------------------------------------------------------------------
extern "C" void kernel_launch(void* const* d_in, const int* in_sizes, int n_in,
                              void* d_out, int out_size, void* d_ws, size_t ws_size,
                              hipStream_t stream)
{
    const float* features = (const float*)d_in[0];
    const int*   captions = (const int*)  d_in[1];
    const int*   sos      = (const int*)  d_in[2];
    const float* emb      = (const float*)d_in[3];
    const float* fc1_W    = (const float*)d_in[4];
    const float* fc1_b    = (const float*)d_in[5];
    const float* attn_Wq  = (const float*)d_in[6];
    const float* attn_bq  = (const float*)d_in[7];
    const float* attn_Wk  = (const float*)d_in[8];
    const float* attn_bk  = (const float*)d_in[9];
    const float* attn_v   = (const float*)d_in[10];
    const float* attn_bv  = (const float*)d_in[11];
    const float* fc0_W    = (const float*)d_in[12];
    const float* fc0_b    = (const float*)d_in[13];
    const float* gru_Wi0  = (const float*)d_in[14];
    const float* gru_Wh0  = (const float*)d_in[15];
    const float* gru_bi0  = (const float*)d_in[16];
    const float* gru_bh0  = (const float*)d_in[17];
    const float* gru_Wi   = (const float*)d_in[18];
    const float* gru_Wh   = (const float*)d_in[19];
    const float* gru_bi   = (const float*)d_in[20];
    const float* gru_bh   = (const float*)d_in[21];
    const float* fc2_W    = (const float*)d_in[22];
    const float* fc2_b    = (const float*)d_in[23];
    float* out = (float*)d_out;

    // workspace layout (floats)
    float* ws = (float*)d_ws;
    size_t o = 0;
    float* kp   = ws + o; o += (size_t)DB * DS * DH;     // keys_proj [B*S, H]
    float* xemb = ws + o; o += (size_t)DB * NSTEP * DE;  // gathered embeddings
    float* xseq = ws + o; o += (size_t)DB * NSTEP * DH;  // fc1 outputs
    float* hbuf = ws + o; o += (size_t)4 * DB * DH;      // 4-layer GRU state
    float* qbuf = ws + o; o += (size_t)DB * DH;
    float* wbuf = ws + o; o += (size_t)DB * DS;
    float* ctxr = ws + o; o += (size_t)DB * DK;
    float* ctxb = ws + o; o += (size_t)DB * DH;
    float* xc   = ws + o; o += (size_t)DB * DK;
    float* gi   = ws + o; o += (size_t)DB * 3 * DH;
    float* gh   = ws + o; o += (size_t)DB * 3 * DH;
    (void)ws_size; (void)in_sizes; (void)n_in; (void)out_size;

    auto gemm = [&](const float* A, int lda, const float* Bm, int ldb, int transB,
                    const float* bias, float* C, int ldc, int M, int N, int Kd) {
        dim3 grid((N + 63) / 64, M / 64);
        gemm_wmma_f32<<<grid, 256, 0, stream>>>(A, lda, Bm, ldb, transB, bias, C, ldc, N, Kd);
    };

    // ---- loop-invariant precompute ----
    embed_gather<<<DB * NSTEP, 128, 0, stream>>>(emb, captions, sos, xemb);
    gemm(xemb, DE, fc1_W, DH, 0, fc1_b, xseq, DH, DB * NSTEP, DH, DE);         // x_seq
    gemm(features, DK, attn_Wk, DH, 0, attn_bk, kp, DH, DB * DS, DH, DK);      // keys_proj
    hipMemsetAsync(hbuf, 0, (size_t)4 * DB * DH * sizeof(float), stream);      // s0 = 0

    // ---- 15 sequential decode steps ----
    for (int t = 0; t < NSTEP; ++t) {
        const float* h3 = hbuf + (size_t)3 * DB * DH;

        // Bahdanau attention
        gemm(h3, DH, attn_Wq, DH, 0, attn_bq, qbuf, DH, DB, DH, DH);
        attn_scores<<<DB, 256, 0, stream>>>(qbuf, kp, attn_v, attn_bv, wbuf);
        ctx_sum<<<DB, 256, 0, stream>>>(wbuf, features, ctxr);
        gemm(ctxr, DK, fc0_W, DH, 0, fc0_b, ctxb, DH, DB, DH, DK);

        // GRU layer 0 (input = [x_t, ctx], 2H)
        concat_x<<<(DB * DK) / 256, 256, 0, stream>>>(xseq, ctxb, xc, t);
        gemm(xc,   DK, gru_Wi0, DK, 1, gru_bi0, gi, 3 * DH, DB, 3 * DH, DK);
        gemm(hbuf, DH, gru_Wh0, DH, 1, gru_bh0, gh, 3 * DH, DB, 3 * DH, DH);
        gru_fuse<<<(DB * DH) / 256, 256, 0, stream>>>(gi, gh, hbuf);

        // GRU layers 1..3
        for (int lr = 1; lr < 4; ++lr) {
            const float* hin = hbuf + (size_t)(lr - 1) * DB * DH;
            float* hcur      = hbuf + (size_t)lr * DB * DH;
            const float* Wi  = gru_Wi + (size_t)(lr - 1) * 3 * DH * DH;
            const float* Wh  = gru_Wh + (size_t)(lr - 1) * 3 * DH * DH;
            gemm(hin,  DH, Wi, DH, 1, gru_bi + (size_t)(lr - 1) * 3 * DH, gi, 3 * DH, DB, 3 * DH, DH);
            gemm(hcur, DH, Wh, DH, 1, gru_bh + (size_t)(lr - 1) * 3 * DH, gh, 3 * DH, DB, 3 * DH, DH);
            gru_fuse<<<(DB * DH) / 256, 256, 0, stream>>>(gi, gh, hcur);
        }

        // output projection into d_out[b, t, :]  (row stride = NSTEP*V)
        gemm(h3, DH, fc2_W, DV, 0, fc2_b, out + (size_t)t * DV, NSTEP * DV, DB, DV, DH);
    }
}